// TAGraph_6665789243857
// MI455X (gfx1250) — compile-verified
//
#include <hip/hip_runtime.h>

typedef __attribute__((ext_vector_type(2))) float v2f;
typedef __attribute__((ext_vector_type(8))) float v8f;

#define N_NODES 100000
#define D 128
#define FANIN 384  // D * (K+1)

// ---------------------------------------------------------------------------
// Zero f1/f2 scratch (float4 stores, 102.4 MB total -> ~4.5us at HBM rate)
// ---------------------------------------------------------------------------
__global__ __launch_bounds__(256) void tag_zero_kernel(float4* __restrict__ p, long n4) {
  long i = (long)blockIdx.x * blockDim.x + threadIdx.x;
  if (i < n4) p[i] = make_float4(0.f, 0.f, 0.f, 0.f);
}

// ---------------------------------------------------------------------------
// Edge-parallel SpMM: y[src[e]] += val[e] * feat[dst[e]]
// One wave (32 lanes) per edge; each lane covers 4 consecutive floats of the
// 128-wide feature row: one global_load_b128 + 4 global_atomic_add_f32.
// Working set (feat + y + edge lists) fits in the 192MB L2, so this is
// L2-gather/atomic throughput bound.
// ---------------------------------------------------------------------------
__global__ __launch_bounds__(256) void tag_spmm_kernel(const int* __restrict__ src,
                                                       const int* __restrict__ dst,
                                                       const float* __restrict__ val,
                                                       const float* __restrict__ feat,
                                                       float* __restrict__ out,
                                                       int E) {
  const int wid  = (blockIdx.x * blockDim.x + threadIdx.x) >> 5;  // wave id == edge id
  const int lane = threadIdx.x & 31;
  if (wid >= E) return;

  const int   s = src[wid];
  const int   d = dst[wid];
  const float v = val[wid];

  const float4 f = *(const float4*)(feat + (long)d * D + lane * 4);
  float* o = out + (long)s * D + lane * 4;
  unsafeAtomicAdd(o + 0, v * f.x);
  unsafeAtomicAdd(o + 1, v * f.y);
  unsafeAtomicAdd(o + 2, v * f.z);
  unsafeAtomicAdd(o + 3, v * f.w);
}

// ---------------------------------------------------------------------------
// out[n, o] = sum_k h[n,k] * W[o,k] + b[o], h = [x | f1 | f2] (never
// materialized; three k-segments with different base pointers).
// Full-f32 WMMA: V_WMMA_F32_16X16X4_F32, 96 MACs per 16x16 tile (K=384).
//
// Per-lane operand layout (ISA 7.12.2, 32-bit 16x4 A / 4x16 B):
//   A: lane L in [0,15]: M=L,    K={0,1} in the two A VGPRs
//      lane L in [16,31]: M=L-16, K={2,3}
//   B: mirrored with N = lane&15.
//   C/D: VGPR r -> row (r + 8*(lane>>4)), col (lane&15).
// ---------------------------------------------------------------------------
__global__ __launch_bounds__(256) void tag_gemm_kernel(const float* __restrict__ x,
                                                       const float* __restrict__ f1,
                                                       const float* __restrict__ f2,
                                                       const float* __restrict__ W,
                                                       const float* __restrict__ bias,
                                                       float* __restrict__ out) {
  const int wave = threadIdx.x >> 5;   // 0..7 -> 16-col tile of the 128 outputs
  const int lane = threadIdx.x & 31;
  const int m    = lane & 15;          // A row / B col within tile
  const int h2   = lane >> 4;          // K sub-offset selector (0 -> K{0,1}, 1 -> K{2,3})
  const int row  = blockIdx.x * 16 + m;
  const int col  = wave * 16 + m;

  const float* __restrict__ segs[3] = { x, f1, f2 };

  v8f acc = {};
#pragma unroll
  for (int s = 0; s < 3; ++s) {
    const float* __restrict__ A = segs[s] + (long)row * D + 2 * h2;
    const float* __restrict__ B = W + (long)col * FANIN + s * D + 2 * h2;
#pragma unroll
    for (int k = 0; k < D; k += 4) {
      v2f a = { A[k], A[k + 1] };
      v2f b = { B[k], B[k + 1] };
      // (neg_a, A, neg_b, B, c_mod, C, reuse_a, reuse_b)
      acc = __builtin_amdgcn_wmma_f32_16x16x4_f32(false, a, false, b,
                                                  (short)0, acc, false, false);
    }
  }

  const float bv = bias[col];
  float* obase = out + (long)blockIdx.x * 16 * D + col;
#pragma unroll
  for (int r = 0; r < 8; ++r) {
    const int orow = r + 8 * h2;
    obase[(long)orow * D] = acc[r] + bv;
  }
}

// ---------------------------------------------------------------------------
extern "C" void kernel_launch(void* const* d_in, const int* in_sizes, int n_in,
                              void* d_out, int out_size, void* d_ws, size_t ws_size,
                              hipStream_t stream) {
  const float* x    = (const float*)d_in[0];
  const int*   esrc = (const int*)d_in[1];
  const int*   edst = (const int*)d_in[2];
  const float* eval = (const float*)d_in[3];
  const float* W    = (const float*)d_in[4];
  const float* b    = (const float*)d_in[5];
  float*       out  = (float*)d_out;
  const int    E    = in_sizes[1];

  // Scratch layout: f1 [N*D], f2 [N*D]  (2 * 51.2 MB)
  float* f1 = (float*)d_ws;
  float* f2 = f1 + (size_t)N_NODES * D;

  const long n4 = (long)2 * N_NODES * D / 4;
  tag_zero_kernel<<<(int)((n4 + 255) / 256), 256, 0, stream>>>((float4*)d_ws, n4);

  const int spmm_blocks = (int)(((long)E * 32 + 255) / 256);  // one wave per edge
  tag_spmm_kernel<<<spmm_blocks, 256, 0, stream>>>(esrc, edst, eval, x,  f1, E);
  tag_spmm_kernel<<<spmm_blocks, 256, 0, stream>>>(esrc, edst, eval, f1, f2, E);

  tag_gemm_kernel<<<N_NODES / 16, 256, 0, stream>>>(x, f1, f2, W, b, out);
}